// MaxCosineSimilarityBlock_12575664243338
// MI455X (gfx1250) — compile-verified
//
#include <hip/hip_runtime.h>
#include <hip/hip_bf16.h>
#include <math.h>

// Problem constants (fixed by setup_inputs)
constexpr int Bn = 16;              // batch
constexpr int Cn = 4;               // channels
constexpr int Ln = 4096;            // sequence length
constexpr int Kn = 256;             // shapelets per channel
constexpr int Sn = 128;             // shapelet length
constexpr int Wn = Ln - Sn + 1;     // 3969 windows
constexpr int NT = (Wn + 15) / 16;  // 249 window tiles of 16
constexpr float EPSV = 1e-8f;

typedef __attribute__((ext_vector_type(2))) float v2f;
typedef __attribute__((ext_vector_type(8))) float v8f;

// ---------------------------------------------------------------------------
// Kernel 1: normalize shapelets along s.  One block (128 threads) per (c,k).
// ---------------------------------------------------------------------------
__global__ __launch_bounds__(128) void shapelet_norm_kernel(
    const float* __restrict__ shp, float* __restrict__ shpn)
{
    __shared__ float red[4];
    const int row = blockIdx.x;          // c*Kn + k, 0..1023
    const int s   = threadIdx.x;         // 0..127
    const float v = shp[row * Sn + s];
    float p = v * v;
    // wave32 butterfly reduction
    #pragma unroll
    for (int off = 16; off >= 1; off >>= 1) p += __shfl_xor(p, off, 32);
    if ((s & 31) == 0) red[s >> 5] = p;
    __syncthreads();
    const float tot = red[0] + red[1] + red[2] + red[3];
    shpn[row * Sn + s] = v / fmaxf(sqrtf(tot), EPSV);
}

// ---------------------------------------------------------------------------
// Kernel 2: sliding-window inverse L2 norm.  One block per (b,c).
// ---------------------------------------------------------------------------
__global__ __launch_bounds__(256) void window_norm_kernel(
    const float* __restrict__ x, float* __restrict__ invn)
{
    const int bc = blockIdx.x;                 // 0..63
    const float* px = x + (size_t)bc * Ln;
    for (int w = threadIdx.x; w < Wn; w += 256) {
        float ss = 0.f;
        #pragma unroll 8
        for (int s = 0; s < Sn; ++s) { const float v = px[w + s]; ss = fmaf(v, v, ss); }
        invn[bc * Wn + w] = 1.0f / fmaxf(sqrtf(ss), EPSV);
    }
}

// ---------------------------------------------------------------------------
// Kernel 3: main correlation via V_WMMA_F32_16X16X4_F32.
// Block = one (b, 16-shapelet tile); 4 waves stride the 249 window tiles.
// A (16x4, M=window, K=s-chunk): lane m=lane&15 holds K={0,1} (lanes 0-15)
//   or K={2,3} (lanes 16-31) -> two f32 loads from x, scaled by invnorm row.
// B (4x16, N=shapelet): pair-packed in LDS so each lane reads one float2.
// C/D 16x16 f32: vgpr r, lane -> (m = r + 8*(lane>=16), n = lane&15).
// ---------------------------------------------------------------------------
__global__ __launch_bounds__(128) void shapelet_wmma_kernel(
    const float* __restrict__ x, const float* __restrict__ shpn,
    const float* __restrict__ invn, float* __restrict__ out)
{
    __shared__ v2f   ldsB[Cn * (Sn / 2) * 16];   // [c][s/2][n] -> 32 KB
    __shared__ float ldsR[4 * 16];

    const int b    = blockIdx.x >> 4;
    const int k0   = (blockIdx.x & 15) << 4;
    const int tid  = threadIdx.x;
    const int wave = tid >> 5;
    const int lane = tid & 31;
    const int n    = lane & 15;
    const int kA   = (lane >> 4) << 1;          // 0 for lanes 0-15, 2 for 16-31

    // Stage normalized shapelet tile: ldsB[((c*64 + s2)*16) + nn] = (shpn[s=2*s2], shpn[s=2*s2+1])
    for (int i = tid; i < Cn * (Sn / 2) * 16; i += 128) {
        const int nn = i & 15;
        const int s2 = (i >> 4) & 63;
        const int c  = i >> 10;
        ldsB[i] = ((const v2f*)shpn)[(size_t)(c * Kn + (k0 + nn)) * (Sn / 2) + s2];
    }
    __syncthreads();

    float rmax = 0.0f;   // relu floor folded into the running max
    for (int t = wave; t < NT; t += 4) {         // wave-uniform trip count
        const int w0 = t << 4;
        const int wv = w0 + n;                   // this lane's window row (m)
        const int wc = wv < Wn ? wv : Wn - 1;    // clamped address, stays in-bounds
        v8f acc = {0.f, 0.f, 0.f, 0.f, 0.f, 0.f, 0.f, 0.f};

        for (int c = 0; c < Cn; ++c) {
            float s_inv = invn[(b * Cn + c) * Wn + wc];
            s_inv = (wv < Wn) ? s_inv : 0.0f;    // zero out padded rows
            const float* px = x + (size_t)(b * Cn + c) * Ln + wc + kA;
            const v2f*   pb = &ldsB[(c * 64) * 16 + n];
            #pragma unroll 4
            for (int s0 = 0; s0 < Sn; s0 += 4) {
                v2f a;
                a.x = px[s0]     * s_inv;        // A: K = kA
                a.y = px[s0 + 1] * s_inv;        // A: K = kA+1
                const v2f bv = pb[((s0 + kA) >> 1) * 16];  // B: rows kA, kA+1
                acc = __builtin_amdgcn_wmma_f32_16x16x4_f32(
                    /*neg_a=*/false, a, /*neg_b=*/false, bv,
                    /*c_mod=*/(short)0, acc, /*reuse_a=*/false, /*reuse_b=*/false);
            }
        }
        #pragma unroll
        for (int r = 0; r < 8; ++r) rmax = fmaxf(rmax, acc[r] * 0.25f);  // /C
    }

    // combine lane halves: lane L holds m=0..7 maxima, lane L+16 holds m=8..15
    rmax = fmaxf(rmax, __shfl_xor(rmax, 16, 32));
    if (lane < 16) ldsR[wave * 16 + n] = rmax;
    __syncthreads();
    if (tid < 16) {
        const float v = fmaxf(fmaxf(ldsR[tid], ldsR[16 + tid]),
                              fmaxf(ldsR[32 + tid], ldsR[48 + tid]));
        out[b * Kn + k0 + tid] = v;   // >= 0, relu satisfied
    }
}

// ---------------------------------------------------------------------------
extern "C" void kernel_launch(void* const* d_in, const int* in_sizes, int n_in,
                              void* d_out, int out_size, void* d_ws, size_t ws_size,
                              hipStream_t stream) {
    const float* x   = (const float*)d_in[0];   // (16,4,4096) f32
    const float* shp = (const float*)d_in[1];   // (4,256,128) f32
    float* out  = (float*)d_out;                // (16,1,256)  f32
    float* shpn = (float*)d_ws;                 // 131072 floats
    float* invn = shpn + Cn * Kn * Sn;          // 254016 floats (total ~1.54 MB ws)

    shapelet_norm_kernel<<<Cn * Kn, Sn, 0, stream>>>(shp, shpn);
    window_norm_kernel<<<Bn * Cn, 256, 0, stream>>>(x, invn);
    shapelet_wmma_kernel<<<Bn * (Kn / 16), 128, 0, stream>>>(x, shpn, invn, out);
}